// LIF_74251394613984
// MI455X (gfx1250) — compile-verified
//
#include <hip/hip_runtime.h>

// LIF scan: out[t] = spike step over membrane recurrence.
// T=100, per-timestep slab = B*F = 262144 floats (1 MB).
// Bandwidth-bound (~210 MB total, ~9 us floor @ 23.3 TB/s).
// Strategy: 16-deep async global->LDS pipeline per wave (ASYNCcnt),
// GVS addressing (scalar base + invariant 32-bit lane offset),
// NT hints on both streams, float4 per lane, wave32.

typedef float v4f __attribute__((ext_vector_type(4)));

#define LIF_T   100
#define PIPE    16          // pipeline depth (timesteps in flight)
#define BLK     256         // threads per block (8 waves, wave32)
#define VEC     4           // floats per thread per timestep
#define CHUNK   (BLK * VEC) // floats per block per timestep = 1024 (4 KB)

// GVS-mode async copy: copies 16 bytes/lane from global sbase plus voffset_bytes to LDS at lds_byte_off.
// Tracked by ASYNCcnt; completes in order per wave. NT: streamed-once data.
__device__ __forceinline__ void async_load_b128(unsigned lds_byte_off,
                                                unsigned voffset_bytes,
                                                const float* sbase) {
    asm volatile("global_load_async_to_lds_b128 %0, %1, %2 th:TH_LOAD_NT"
                 :: "v"(lds_byte_off), "v"(voffset_bytes), "s"(sbase)
                 : "memory");
}

// Wait until at most `imm` async ops remain outstanding.
#define ASYNC_WAIT(imm) asm volatile("s_wait_asynccnt " #imm ::: "memory")

__global__ void __launch_bounds__(BLK)
lif_scan_kernel(const float* __restrict__ in, float* __restrict__ out,
                int n_per_t) {
    __shared__ v4f buf[PIPE][BLK];   // 16 x 4 KB = 64 KB LDS

    const int tid = threadIdx.x;
    // Loop-invariant per-lane byte offset within a timestep slab.
    const unsigned voff = (unsigned)(blockIdx.x * CHUNK + tid * VEC) * 4u;
    const long long e0  = (long long)blockIdx.x * CHUNK + (long long)tid * VEC;

    // Prologue: fill the pipeline with timesteps 0..PIPE-1 (scalar bases only).
    #pragma unroll
    for (int k = 0; k < PIPE; ++k) {
        unsigned lds = (unsigned)(uintptr_t)&buf[k][tid]; // low 32 bits = LDS byte offset
        async_load_b128(lds, voff, in + (long long)k * n_per_t);
    }

    v4f mem = (v4f)(0.0f);
    v4f spk = (v4f)(0.0f);

    for (int t = 0; t < LIF_T; ++t) {
        // Steady state: PIPE loads in flight; in-order completion means
        // asynccnt <= PIPE-1 guarantees load t has landed. Tail: drain fully.
        if (t <= LIF_T - 1 - PIPE) {
            ASYNC_WAIT(15);
        } else {
            ASYNC_WAIT(0);
        }

        v4f x = buf[t & (PIPE - 1)][tid];   // ds_load_b128

        #pragma unroll
        for (int i = 0; i < VEC; ++i) {
            // mem = v_decay * mem * (1 - spk) + x ; spk = (mem > v_th)
            // (mem/0.5 - 1 > 0  <=>  mem > 0.5, exact in fp32)
            mem[i] = 0.2f * mem[i] * (1.0f - spk[i]) + x[i];
            spk[i] = (mem[i] > 0.5f) ? 1.0f : 0.0f;
        }

        // Streaming write-once output: non-temporal 128-bit store.
        __builtin_nontemporal_store(spk,
            (v4f*)(out + (long long)t * n_per_t + e0));

        // Refill the slot we just freed with timestep t+PIPE.
        const int nxt = t + PIPE;
        if (nxt < LIF_T) {
            unsigned lds = (unsigned)(uintptr_t)&buf[nxt & (PIPE - 1)][tid];
            async_load_b128(lds, voff, in + (long long)nxt * n_per_t);
        }
    }
}

extern "C" void kernel_launch(void* const* d_in, const int* in_sizes, int n_in,
                              void* d_out, int out_size, void* d_ws, size_t ws_size,
                              hipStream_t stream) {
    const float* in  = (const float*)d_in[0];
    float*       out = (float*)d_out;

    const int total   = in_sizes[0];        // T * B * F
    const int n_per_t = total / LIF_T;      // B * F = 262144
    const int blocks  = n_per_t / CHUNK;    // 256 workgroups (2048 waves)

    lif_scan_kernel<<<blocks, BLK, 0, stream>>>(in, out, n_per_t);
}